// match_GRU_75539884802387
// MI455X (gfx1250) — compile-verified
//
#include <hip/hip_runtime.h>
#include <math.h>

typedef __attribute__((ext_vector_type(2))) float v2f;
typedef __attribute__((ext_vector_type(8))) float v8f;

constexpr int Bc = 4, LKc = 512, LQc = 128, Dc = 256, Hc = 256, Lc = 3;

// ---------------------------------------------------------------------------
// CDNA5 async memory->LDS helpers (ASYNCcnt-tracked, ISA ch.10 / 15.18.3).
// Flat pointers to LDS hold the LDS byte offset in their low 32 bits
// (ISA 10.2 aperture mapping).
// ---------------------------------------------------------------------------
__device__ __forceinline__ unsigned lds_off(const void* p) {
  return (unsigned)(unsigned long long)p;
}
__device__ __forceinline__ void async_ld_b128(unsigned ldsaddr, const void* gaddr) {
  asm volatile("global_load_async_to_lds_b128 %0, %1, off"
               :: "v"(ldsaddr), "v"(gaddr) : "memory");
}
__device__ __forceinline__ void async_ld_b32(unsigned ldsaddr, const void* gaddr) {
  asm volatile("global_load_async_to_lds_b32 %0, %1, off"
               :: "v"(ldsaddr), "v"(gaddr) : "memory");
}
__device__ __forceinline__ void wait_asynccnt0() {
  asm volatile("s_wait_asynccnt 0x0" ::: "memory");
}

// ---------------------------------------------------------------------------
// WMMA f32 16x16x4 fragment layouts (ISA 7.12.2):
//   A: lane l -> m = l&15, VGPR j holds K = j + 2*(l>>4)
//   B: VGPR j lane l -> (k = j + 2*(l>>4), n = l&15)
//   C/D: VGPR r lane l -> (m = r + 8*(l>>4), n = l&15)
// ---------------------------------------------------------------------------
#define WMMA_F32(a, b, c) \
  __builtin_amdgcn_wmma_f32_16x16x4_f32(false, (a), false, (b), (short)0, (c), false, false)

// single-tile: C += A[16 x K] * W[16 x K]^T  (W row-major N x K)
__device__ __forceinline__ v8f wmma_acc_awt(const float* __restrict__ A, int lda,
                                            const float* __restrict__ W, int ldw,
                                            int K, v8f acc) {
  const int lane = threadIdx.x & 31;
  const int half = lane >> 4;
  const int l16  = lane & 15;
  const float* arow = A + l16 * lda + 2 * half;
  const float* wrow = W + l16 * ldw + 2 * half;
  for (int k = 0; k < K; k += 4) {
    v2f a, b;
    a.x = arow[k];     a.y = arow[k + 1];
    b.x = wrow[k];     b.y = wrow[k + 1];
    acc = WMMA_F32(a, b, acc);
  }
  return acc;
}

// ---------------------------------------------------------------------------
// Kernel 1: C[M x 256] = A[M x 256] * W[256 x 256]^T   (projections)
// Each wave: 1 M-tile x 4 N-tiles; A-fragment reused across 4 WMMAs.
// ---------------------------------------------------------------------------
__global__ void proj_kernel(const float* __restrict__ A, const float* __restrict__ W,
                            float* __restrict__ C, int M) {
  const int nJobs = (M / 16) * 4;            // 4 n-groups of 64 columns
  int job = blockIdx.x * 4 + (threadIdx.x >> 5);
  if (job >= nJobs) return;                  // wave-uniform guard
  int mt = job >> 2, ng = job & 3;
  const int lane = threadIdx.x & 31, half = lane >> 4, l16 = lane & 15;
  const float* arow = A + (mt * 16 + l16) * Dc + 2 * half;
  const float* w0 = W + (ng * 64 +  0 + l16) * Dc + 2 * half;
  const float* w1 = W + (ng * 64 + 16 + l16) * Dc + 2 * half;
  const float* w2 = W + (ng * 64 + 32 + l16) * Dc + 2 * half;
  const float* w3 = W + (ng * 64 + 48 + l16) * Dc + 2 * half;
  v8f acc0 = {0.f,0.f,0.f,0.f,0.f,0.f,0.f,0.f};
  v8f acc1 = acc0, acc2 = acc0, acc3 = acc0;
  for (int k = 0; k < Dc; k += 4) {
    v2f a, b0, b1, b2, b3;
    a.x  = arow[k]; a.y  = arow[k + 1];
    b0.x = w0[k];   b0.y = w0[k + 1];
    b1.x = w1[k];   b1.y = w1[k + 1];
    b2.x = w2[k];   b2.y = w2[k + 1];
    b3.x = w3[k];   b3.y = w3[k + 1];
    acc0 = WMMA_F32(a, b0, acc0);
    acc1 = WMMA_F32(a, b1, acc1);
    acc2 = WMMA_F32(a, b2, acc2);
    acc3 = WMMA_F32(a, b3, acc3);
  }
  int nb = ng * 64 + l16;
#define PST(g, r) C[(mt * 16 + (r) + 8 * half) * Hc + nb + (g) * 16] = acc##g[r];
#define PSTROW(g) PST(g,0) PST(g,1) PST(g,2) PST(g,3) PST(g,4) PST(g,5) PST(g,6) PST(g,7)
  PSTROW(0) PSTROW(1) PSTROW(2) PSTROW(3)
#undef PSTROW
#undef PST
}

// ---------------------------------------------------------------------------
// Kernel 2: E[b,q,k] = exp( sum_h w_v[h] * tanh(Q[b,q,h] + K[b,k,h]) )
// grid = (LK/16, LQ/16, B), block = 256.  Async memory->LDS staging.
// ---------------------------------------------------------------------------
__global__ void score_kernel(const float* __restrict__ Qb, const float* __restrict__ Kb,
                             const float* __restrict__ w_v, float* __restrict__ E) {
  constexpr int LD = 260;                    // padded row stride (floats)
  __shared__ float Qs[16 * LD];
  __shared__ float Ks[16 * LD];
  __shared__ float wv[256];
  const int tid = threadIdx.x;
  const int k0 = blockIdx.x * 16, q0 = blockIdx.y * 16, b = blockIdx.z;

  // stage 16x256 Q and K tiles with async b128 (16B) transfers: 1024 chunks each
  for (int i = tid; i < 1024; i += 256) {
    int r = i >> 6;                          // row 0..15
    int c = (i & 63) << 2;                   // col (floats), multiple of 4
    async_ld_b128(lds_off(&Qs[r * LD + c]), &Qb[(b * LQc + q0 + r) * Hc + c]);
    async_ld_b128(lds_off(&Ks[r * LD + c]), &Kb[(b * LKc + k0 + r) * Hc + c]);
  }
  wv[tid] = w_v[tid];
  wait_asynccnt0();
  __syncthreads();

  const int tq = tid >> 4, tk = tid & 15;
  float s = 0.f;
  const float* qr = &Qs[tq * LD];
  const float* kr = &Ks[tk * LD];
#pragma unroll 4
  for (int h = 0; h < 256; ++h)
    s += wv[h] * tanhf(qr[h] + kr[h]);
  E[(b * LQc + q0 + tq) * LKc + k0 + tk] = expf(s);
}

// ---------------------------------------------------------------------------
// Kernel 3: inv[q] = 1 / sum_{b,k} E[b,q,k]   (global cross-batch normalizer)
// ---------------------------------------------------------------------------
__global__ void denom_kernel(const float* __restrict__ E, float* __restrict__ inv) {
  __shared__ float red[256];
  const int q = blockIdx.x, tid = threadIdx.x;
  float s = 0.f;
  for (int i = tid; i < Bc * LKc; i += 256) {
    int b = i / LKc, k = i % LKc;
    s += E[(b * LQc + q) * LKc + k];
  }
  red[tid] = s;
  __syncthreads();
  for (int off = 128; off > 0; off >>= 1) {
    if (tid < off) red[tid] += red[tid + off];
    __syncthreads();
  }
  if (tid == 0) inv[q] = 1.0f / red[0];
}

// ---------------------------------------------------------------------------
// Kernel 4: Uc[b,q,d] = queries[b,q,d] + inv[q] * sum_k E[b,q,k] * keys[b,k,d]
// grid = (8, B), block = 128.  Each wave: 1 q-tile x 4 d-tiles (A reuse).
// ---------------------------------------------------------------------------
__global__ void attn_kernel(const float* __restrict__ E, const float* __restrict__ keys,
                            const float* __restrict__ inv, const float* __restrict__ queries,
                            float* __restrict__ Uc) {
  const int b = blockIdx.y;
  int job = blockIdx.x * 4 + (threadIdx.x >> 5);   // 32 jobs: mt(8) x ng(4)
  int mt = job >> 2, ng = job & 3;
  const int lane = threadIdx.x & 31, half = lane >> 4, l16 = lane & 15;
  const float* arow  = E + (size_t)(b * LQc + mt * 16 + l16) * LKc + 2 * half;
  const float* bbase = keys + (size_t)b * LKc * Dc + (2 * half) * Dc + ng * 64 + l16;
  v8f acc0 = {0.f,0.f,0.f,0.f,0.f,0.f,0.f,0.f};
  v8f acc1 = acc0, acc2 = acc0, acc3 = acc0;
  for (int k = 0; k < LKc; k += 4) {
    v2f a, b0, b1, b2, b3;
    a.x  = arow[k];                 a.y  = arow[k + 1];
    b0.x = bbase[k * Dc +  0];      b0.y = bbase[(k + 1) * Dc +  0];
    b1.x = bbase[k * Dc + 16];      b1.y = bbase[(k + 1) * Dc + 16];
    b2.x = bbase[k * Dc + 32];      b2.y = bbase[(k + 1) * Dc + 32];
    b3.x = bbase[k * Dc + 48];      b3.y = bbase[(k + 1) * Dc + 48];
    acc0 = WMMA_F32(a, b0, acc0);
    acc1 = WMMA_F32(a, b1, acc1);
    acc2 = WMMA_F32(a, b2, acc2);
    acc3 = WMMA_F32(a, b3, acc3);
  }
#define AST(g, r) { int q = mt * 16 + (r) + 8 * half;                          \
                    int idx = (b * LQc + q) * Dc + ng * 64 + (g) * 16 + l16;   \
                    Uc[idx] = queries[idx] + acc##g[r] * inv[q]; }
#define ASTROW(g) AST(g,0) AST(g,1) AST(g,2) AST(g,3) AST(g,4) AST(g,5) AST(g,6) AST(g,7)
  ASTROW(0) ASTROW(1) ASTROW(2) ASTROW(3)
#undef ASTROW
#undef AST
}

// ---------------------------------------------------------------------------
// Kernel 5: X0[m,n] = sigmoid((Uc @ Wg^T)[m,n] + bg[n]) * Uc[m,n], m in [0,512)
// Each wave: 1 m-tile x 4 n-tiles.
// ---------------------------------------------------------------------------
__global__ void gate_kernel(const float* __restrict__ Uc, const float* __restrict__ Wg,
                            const float* __restrict__ bg, float* __restrict__ X0) {
  int job = blockIdx.x * 4 + (threadIdx.x >> 5);   // 128 jobs: mt(32) x ng(4)
  int mt = job >> 2, ng = job & 3;
  const int lane = threadIdx.x & 31, half = lane >> 4, l16 = lane & 15;
  const float* arow = Uc + (mt * 16 + l16) * Hc + 2 * half;
  const float* w0 = Wg + (ng * 64 +  0 + l16) * Hc + 2 * half;
  const float* w1 = Wg + (ng * 64 + 16 + l16) * Hc + 2 * half;
  const float* w2 = Wg + (ng * 64 + 32 + l16) * Hc + 2 * half;
  const float* w3 = Wg + (ng * 64 + 48 + l16) * Hc + 2 * half;
  v8f acc0 = {0.f,0.f,0.f,0.f,0.f,0.f,0.f,0.f};
  v8f acc1 = acc0, acc2 = acc0, acc3 = acc0;
  for (int k = 0; k < Hc; k += 4) {
    v2f a, b0, b1, b2, b3;
    a.x  = arow[k]; a.y  = arow[k + 1];
    b0.x = w0[k];   b0.y = w0[k + 1];
    b1.x = w1[k];   b1.y = w1[k + 1];
    b2.x = w2[k];   b2.y = w2[k + 1];
    b3.x = w3[k];   b3.y = w3[k + 1];
    acc0 = WMMA_F32(a, b0, acc0);
    acc1 = WMMA_F32(a, b1, acc1);
    acc2 = WMMA_F32(a, b2, acc2);
    acc3 = WMMA_F32(a, b3, acc3);
  }
#define GST(g, r) { int m = mt * 16 + (r) + 8 * half;                          \
                    int n = ng * 64 + (g) * 16 + l16;                          \
                    float tv = acc##g[r] + bg[n];                              \
                    float gg = 1.f / (1.f + expf(-tv));                        \
                    X0[m * Hc + n] = gg * Uc[m * Hc + n]; }
#define GSTROW(g) GST(g,0) GST(g,1) GST(g,2) GST(g,3) GST(g,4) GST(g,5) GST(g,6) GST(g,7)
  GSTROW(0) GSTROW(1) GSTROW(2) GSTROW(3)
#undef GSTROW
#undef GST
}

// ---------------------------------------------------------------------------
// Kernel 6: 3-layer GRU, h0 = 0.  Single persistent workgroup (32 waves).
// Per step: 96 WMMA tile-jobs (48 gi + 48 gh), 3 per wave; gates by 1024 thr.
// x_t staged to LDS via async load; B=4 rows padded to M=16 with zero rows.
// ---------------------------------------------------------------------------
__global__ __launch_bounds__(1024) void gru_kernel(
    const float* __restrict__ x0,
    const float* __restrict__ Wih, const float* __restrict__ Whh,
    const float* __restrict__ bih, const float* __restrict__ bhh,
    float* __restrict__ y0, float* __restrict__ y1, float* __restrict__ out) {
  constexpr int LD = 260;                     // padded LDS row stride
  __shared__ float xs[16 * LD];
  __shared__ float hs[16 * LD];
  __shared__ float gi[4 * 768];
  __shared__ float gh[4 * 768];

  const int tid = threadIdx.x;
  const int lane = tid & 31, half = lane >> 4, l16 = lane & 15;
  const int wvid = tid >> 5;                  // 0..31
  const int bb = tid >> 8, jj = tid & 255;    // (batch, feature) phases

  // zero pad rows 4..15 once (only cols < 256 are ever read by WMMA)
  for (int i = tid; i < 12 * 256; i += 1024) {
    int r = 4 + i / 256, c = i & 255;
    xs[r * LD + c] = 0.f;
    hs[r * LD + c] = 0.f;
  }

  for (int l = 0; l < Lc; ++l) {
    const float* xin  = (l == 0) ? x0 : ((l == 1) ? y0 : y1);
    float*       yout = (l == 0) ? y0 : ((l == 1) ? y1 : out);
    const float* Wih_l = Wih + l * 3 * Hc * Hc;
    const float* Whh_l = Whh + l * 3 * Hc * Hc;
    const float* bih_l = bih + l * 3 * Hc;
    const float* bhh_l = bhh + l * 3 * Hc;

    hs[bb * LD + jj] = 0.f;                   // h0 = 0
    __syncthreads();

    for (int t = 0; t < LQc; ++t) {
      // async stage x_t rows 0..3 (4 KB) directly into LDS
      async_ld_b32(lds_off(&xs[bb * LD + jj]), &xin[(bb * LQc + t) * Hc + jj]);
      wait_asynccnt0();
      __syncthreads();

#pragma unroll
      for (int job_i = 0; job_i < 3; ++job_i) {
        int job = wvid + job_i * 32;          // 0..95, wave-uniform
        bool is_gh = job >= 48;
        int nt = is_gh ? (job - 48) : job;    // 0..47 (N tile of 3H=768)
        const float* Asrc = is_gh ? hs : xs;
        const float* Wsrc = (is_gh ? Whh_l : Wih_l) + nt * 16 * Hc;
        v8f acc = {0.f,0.f,0.f,0.f,0.f,0.f,0.f,0.f};
        acc = wmma_acc_awt(Asrc, LD, Wsrc, Hc, Hc, acc);
        float* gdst = is_gh ? gh : gi;
        const float* bias = is_gh ? bhh_l : bih_l;
        if (half == 0) {                      // rows m = r (only m<4 valid)
          int n = nt * 16 + l16;
          float bv = bias[n];
          gdst[0 * 768 + n] = acc[0] + bv;
          gdst[1 * 768 + n] = acc[1] + bv;
          gdst[2 * 768 + n] = acc[2] + bv;
          gdst[3 * 768 + n] = acc[3] + bv;
        }
      }
      __syncthreads();

      {
        float ir  = gi[bb * 768 + jj],        hr = gh[bb * 768 + jj];
        float iz  = gi[bb * 768 + 256 + jj],  hz = gh[bb * 768 + 256 + jj];
        float in_ = gi[bb * 768 + 512 + jj],  hn = gh[bb * 768 + 512 + jj];
        float r = 1.f / (1.f + expf(-(ir + hr)));
        float z = 1.f / (1.f + expf(-(iz + hz)));
        float n = tanhf(in_ + r * hn);
        float hnew = (1.f - z) * n + z * hs[bb * LD + jj];
        hs[bb * LD + jj] = hnew;
        yout[(bb * LQc + t) * Hc + jj] = hnew;
      }
      __syncthreads();
    }
    __syncthreads();
  }
}

// ---------------------------------------------------------------------------
extern "C" void kernel_launch(void* const* d_in, const int* in_sizes, int n_in,
                              void* d_out, int out_size, void* d_ws, size_t ws_size,
                              hipStream_t stream) {
  (void)in_sizes; (void)n_in; (void)out_size; (void)ws_size;
  const float* keys    = (const float*)d_in[0];
  const float* queries = (const float*)d_in[1];
  const float* W_k     = (const float*)d_in[2];
  const float* W_q     = (const float*)d_in[3];
  const float* w_v     = (const float*)d_in[4];
  const float* W_g_w   = (const float*)d_in[5];
  const float* W_g_b   = (const float*)d_in[6];
  const float* gWih    = (const float*)d_in[7];
  const float* gWhh    = (const float*)d_in[8];
  const float* gbih    = (const float*)d_in[9];
  const float* gbhh    = (const float*)d_in[10];
  float* out = (float*)d_out;

  float* ws  = (float*)d_ws;
  float* Kb  = ws;                       // [B*LK, H]  524288
  float* Qb  = Kb  + Bc * LKc * Hc;      // [B*LQ, H]  131072
  float* Eb  = Qb  + Bc * LQc * Hc;      // [B, LQ, LK] 262144
  float* inv = Eb  + Bc * LQc * LKc;     // [LQ] (padded to 256)
  float* Uc  = inv + 256;                // [B*LQ, H]  131072
  float* X0  = Uc  + Bc * LQc * Hc;      // [B*LQ, H]
  float* Y0  = X0  + Bc * LQc * Hc;      // [B*LQ, H]
  float* Y1  = Y0  + Bc * LQc * Hc;      // [B*LQ, H]   total ~5.8 MB

  // K = keys @ W_k^T ; Q = queries @ W_q^T    (fp32 WMMA, 4 n-tiles/wave)
  proj_kernel<<<128, 128, 0, stream>>>(keys,    W_k, Kb, Bc * LKc);
  proj_kernel<<< 32, 128, 0, stream>>>(queries, W_q, Qb, Bc * LQc);
  // e = exp(tanh-score), async LDS staging
  score_kernel<<<dim3(LKc / 16, LQc / 16, Bc), 256, 0, stream>>>(Qb, Kb, w_v, Eb);
  // global (cross-batch, cross-key) normalizer per q
  denom_kernel<<<LQc, 256, 0, stream>>>(Eb, inv);
  // u_c = queries + att @ keys
  attn_kernel<<<dim3(8, Bc), 128, 0, stream>>>(Eb, keys, inv, queries, Uc);
  // x0 = sigmoid(u_c @ Wg^T + bg) * u_c
  gate_kernel<<<32, 128, 0, stream>>>(Uc, W_g_w, W_g_b, X0);
  // 3-layer GRU -> d_out
  gru_kernel<<<1, 1024, 0, stream>>>(X0, gWih, gWhh, gbih, gbhh, Y0, Y1, out);
}